// EdgesToNodes_57329223467233
// MI455X (gfx1250) — compile-verified
//
#include <hip/hip_runtime.h>
#include <hip/hip_bf16.h>

// ---------------------------------------------------------------------------
// EdgesToNodes: wxe = W[:,None,:]*xe ; acc1 = segsum(wxe, dst) ; acc2 = segsum(wxe, src)
//               out = acc1 @ (M1+M2)/2 + acc2 @ (M2-M1)/2       (fp32 everywhere)
// Phase 1: streaming + L2 float atomics (bandwidth bound, ~416 MB reads)
// Phase 2: V_WMMA_F32_16X16X4_F32 over [150000 x 32] @ [32 x 32]
// ---------------------------------------------------------------------------

#define N_EDGES 800000
#define N_NODES 50000
#define MID 3
#define NVEC 32
#define ROWS (N_NODES * MID)          // 150000 accumulator rows of 32 floats
#define ROW_TILES (ROWS / 16)         // 9375, exact

typedef float v2f __attribute__((ext_vector_type(2)));
typedef float v8f __attribute__((ext_vector_type(8)));

// ---------------------------------------------------------------------------
// Kernel 0: zero the two node accumulators (must run every launch: graph
// replays accumulate via atomics and the harness doesn't re-poison).
// 2 * 150000 * 32 floats = 9.6M floats = 2.4M float4  ->  9375 blocks x 256.
// ---------------------------------------------------------------------------
__global__ void zero_acc_kernel(float4* __restrict__ p) {
    int i = blockIdx.x * blockDim.x + threadIdx.x;   // exactly covers 2.4M
    p[i] = make_float4(0.f, 0.f, 0.f, 0.f);
}

// ---------------------------------------------------------------------------
// Kernel 1: per-edge weighted feature + dual scatter-add.
// Work item = one float4 chunk of wxe: E * 3 * 8 = 19.2M chunks -> 75000 x 256.
// chunk c: e = c/24, r = c%24 (r = mid*8 + v4). Coalesced float4 loads of xe;
// W float4 rebroadcast across the 3 mids from L1; indices broadcast from L1.
// 8 no-return global_atomic_add_f32 per thread, landing in L2 (38.4MB accs).
// ---------------------------------------------------------------------------
__global__ void edge_scatter_kernel(const float4* __restrict__ xe4,
                                    const float4* __restrict__ W4,
                                    const int*    __restrict__ src,
                                    const int*    __restrict__ dst,
                                    float* __restrict__ acc1,
                                    float* __restrict__ acc2) {
    const int c = blockIdx.x * blockDim.x + threadIdx.x;  // < 19,200,000 exactly
    const int e = c / 24;
    const int r = c - e * 24;       // mid*8 + v4
    const int v4 = r & 7;

    // CDNA5 prefetch (global_prefetch_b8): pull the xe stream ahead.
    if (c + 4096 < N_EDGES * 24)
        __builtin_prefetch(xe4 + c + 4096, 0, 0);

    const float4 x = xe4[c];                 // xe[e, mid, v4*4 .. +3]
    const float4 w = W4[e * 8 + v4];         // W [e, v4*4 .. +3]
    const float4 v = make_float4(x.x * w.x, x.y * w.y, x.z * w.z, x.w * w.w);

    const int d = dst[e];
    const int s = src[e];
    float* p1 = acc1 + (size_t)d * 96 + r * 4;
    float* p2 = acc2 + (size_t)s * 96 + r * 4;

    (void)__hip_atomic_fetch_add(p1 + 0, v.x, __ATOMIC_RELAXED, __HIP_MEMORY_SCOPE_AGENT);
    (void)__hip_atomic_fetch_add(p1 + 1, v.y, __ATOMIC_RELAXED, __HIP_MEMORY_SCOPE_AGENT);
    (void)__hip_atomic_fetch_add(p1 + 2, v.z, __ATOMIC_RELAXED, __HIP_MEMORY_SCOPE_AGENT);
    (void)__hip_atomic_fetch_add(p1 + 3, v.w, __ATOMIC_RELAXED, __HIP_MEMORY_SCOPE_AGENT);
    (void)__hip_atomic_fetch_add(p2 + 0, v.x, __ATOMIC_RELAXED, __HIP_MEMORY_SCOPE_AGENT);
    (void)__hip_atomic_fetch_add(p2 + 1, v.y, __ATOMIC_RELAXED, __HIP_MEMORY_SCOPE_AGENT);
    (void)__hip_atomic_fetch_add(p2 + 2, v.z, __ATOMIC_RELAXED, __HIP_MEMORY_SCOPE_AGENT);
    (void)__hip_atomic_fetch_add(p2 + 3, v.w, __ATOMIC_RELAXED, __HIP_MEMORY_SCOPE_AGENT);
}

// ---------------------------------------------------------------------------
// Kernel 2: node transform via V_WMMA_F32_16X16X4_F32.
// One wave per 16-row tile of the [150000 x 32] accumulators; two 16x16
// accumulators (left/right column tiles of the 32-wide output). K=32 chained
// as 8 steps of K=4; per step 4 WMMAs (a1@Ma, a2@Mb for each column tile)
// where Ma=(M1+M2)/2, Mb=(M2-M1)/2 are formed in-register.
//
// FP32 WMMA VGPR layouts (ISA 7.12.2):
//   A 16x4 : lane<16 -> (M=lane, K=k0+{0,1}) ; lane>=16 -> (M=lane-16, K=k0+{2,3})
//   B 4x16 : lane<16 -> (K=k0+{0,1}, N=lane) ; lane>=16 -> (K=k0+{2,3}, N=lane-16)
//   C/D    : VGPR r  -> (M=r + (lane>=16 ? 8:0), N=lane&15)
//
// 9375 tiles = 3125 blocks x 96 threads (3 waves/block), no partial waves
// -> EXEC is all-ones for every WMMA (hardware requirement).
// ---------------------------------------------------------------------------
__global__ void node_wmma_kernel(const float* __restrict__ acc1,
                                 const float* __restrict__ acc2,
                                 const float* __restrict__ M1,
                                 const float* __restrict__ M2,
                                 float* __restrict__ out) {
    const int lane  = threadIdx.x & 31;
    const int wave  = threadIdx.x >> 5;                 // 0..2
    const int tile  = blockIdx.x * 3 + wave;            // 0..9374, exact cover
    const int row0  = tile * 16;

    const int mrow  = lane & 15;    // A: M index / B,C,D: N index within tile
    const int khalf = lane >> 4;    // 0 -> K pair {0,1}; 1 -> K pair {2,3}

    const float* a1base = acc1 + (size_t)(row0 + mrow) * NVEC;
    const float* a2base = acc2 + (size_t)(row0 + mrow) * NVEC;

    v8f dL = {0.f, 0.f, 0.f, 0.f, 0.f, 0.f, 0.f, 0.f};
    v8f dR = {0.f, 0.f, 0.f, 0.f, 0.f, 0.f, 0.f, 0.f};

#pragma unroll
    for (int kk = 0; kk < 8; ++kk) {
        const int k0 = kk * 4 + khalf * 2;

        // A fragments: two K-adjacent f32 per lane (8B-aligned: k0 is even).
        const v2f a1 = *(const v2f*)(a1base + k0);
        const v2f a2 = *(const v2f*)(a2base + k0);

        // B fragments: rows k0, k0+1 of the combined matrices; columns
        // mrow (left tile) and 16+mrow (right tile). Tiny, L2/L1-resident.
        const float l1a = M1[(size_t)k0 * 32 + mrow];
        const float l2a = M2[(size_t)k0 * 32 + mrow];
        const float l1b = M1[(size_t)(k0 + 1) * 32 + mrow];
        const float l2b = M2[(size_t)(k0 + 1) * 32 + mrow];
        const float r1a = M1[(size_t)k0 * 32 + 16 + mrow];
        const float r2a = M2[(size_t)k0 * 32 + 16 + mrow];
        const float r1b = M1[(size_t)(k0 + 1) * 32 + 16 + mrow];
        const float r2b = M2[(size_t)(k0 + 1) * 32 + 16 + mrow];

        const v2f MaL = {0.5f * (l1a + l2a), 0.5f * (l1b + l2b)};
        const v2f MbL = {0.5f * (l2a - l1a), 0.5f * (l2b - l1b)};
        const v2f MaR = {0.5f * (r1a + r2a), 0.5f * (r1b + r2b)};
        const v2f MbR = {0.5f * (r2a - r1a), 0.5f * (r2b - r1b)};

        dL = __builtin_amdgcn_wmma_f32_16x16x4_f32(false, a1, false, MaL,
                                                   (short)0, dL, false, false);
        dL = __builtin_amdgcn_wmma_f32_16x16x4_f32(false, a2, false, MbL,
                                                   (short)0, dL, false, false);
        dR = __builtin_amdgcn_wmma_f32_16x16x4_f32(false, a1, false, MaR,
                                                   (short)0, dR, false, false);
        dR = __builtin_amdgcn_wmma_f32_16x16x4_f32(false, a2, false, MbR,
                                                   (short)0, dR, false, false);
    }

    // Store D tiles per the C/D layout.
    const int outRow0 = row0 + khalf * 8;
#pragma unroll
    for (int r = 0; r < 8; ++r) {
        out[(size_t)(outRow0 + r) * 32 + mrow]      = dL[r];
        out[(size_t)(outRow0 + r) * 32 + 16 + mrow] = dR[r];
    }
}

// ---------------------------------------------------------------------------
// Launch. Inputs (setup_inputs order): xe, W, M1, M2, xe_src, xe_dst, n_nodes.
// Workspace: acc1 (4.8M f32) | acc2 (4.8M f32) = 38.4 MB.
// ---------------------------------------------------------------------------
extern "C" void kernel_launch(void* const* d_in, const int* in_sizes, int n_in,
                              void* d_out, int out_size, void* d_ws, size_t ws_size,
                              hipStream_t stream) {
    const float4* xe4 = (const float4*)d_in[0];
    const float4* W4  = (const float4*)d_in[1];
    const float*  M1  = (const float*)d_in[2];
    const float*  M2  = (const float*)d_in[3];
    const int*    src = (const int*)d_in[4];
    const int*    dst = (const int*)d_in[5];
    float* out = (float*)d_out;

    float* acc1 = (float*)d_ws;
    float* acc2 = acc1 + (size_t)ROWS * NVEC;

    // 0) zero accumulators: 2*150000*32/4 = 2,400,000 float4 -> 9375 x 256 exact
    zero_acc_kernel<<<9375, 256, 0, stream>>>((float4*)d_ws);

    // 1) edge scatter: 800000*24 = 19,200,000 chunks -> 75000 x 256 exact
    edge_scatter_kernel<<<75000, 256, 0, stream>>>(xe4, W4, src, dst, acc1, acc2);

    // 2) WMMA transform: 9375 wave-tiles -> 3125 blocks x 96 threads exact
    node_wmma_kernel<<<3125, 96, 0, stream>>>(acc1, acc2, M1, M2, out);
}